// TemporalAttention_14723147890815
// MI455X (gfx1250) — compile-verified
//
#include <hip/hip_runtime.h>
#include <math.h>

// Problem shape (fixed by the reference)
#define B_   2048
#define L_   512
#define H_   128
#define XPAD 132   // LDS row stride in floats: 132*4=528B -> 16B aligned, bank step 4

typedef __attribute__((ext_vector_type(2))) float    v2f;
typedef __attribute__((ext_vector_type(8))) float    v8f;
typedef __attribute__((ext_vector_type(4))) unsigned v4u;
typedef __attribute__((ext_vector_type(8))) int      v8i;
typedef __attribute__((ext_vector_type(4))) int      v4i;

// ---------------------------------------------------------------------------
// Kernel 1: qk[b,:] = (last_hidden[b,:] @ Wq^T + bq) @ Wk
//           qb[b]   = (last_hidden[b,:] @ Wq^T + bq) . bk
// One block = 16 batch rows, 8 waves; wave w owns output N-tile [w*16, w*16+16).
// Uses V_WMMA_F32_16X16X4_F32, K consumed 4 at a time (32 steps for K=128).
// ---------------------------------------------------------------------------
__global__ __launch_bounds__(256) void ta_qk_kernel(
    const float* __restrict__ last_hidden,   // [B,H]
    const float* __restrict__ Wq,            // [H,H]
    const float* __restrict__ bq,            // [H]
    const float* __restrict__ Wk,            // [H,H]
    const float* __restrict__ bk,            // [H]
    float* __restrict__ qk_out,              // [B,H]  (workspace)
    float* __restrict__ qb_out)              // [B]    (workspace)
{
  __shared__ float q_lds[16][130];           // query tile, padded vs bank conflicts

  const int m0   = blockIdx.x * 16;          // batch rows handled by this block
  const int lane = threadIdx.x & 31;
  const int wave = threadIdx.x >> 5;
  const int r16  = lane & 15;                // tile row / col index for this lane
  const int kh   = (lane >> 4) << 1;         // 0 (lanes 0-15) or 2 (lanes 16-31)
  const int mb   = (lane >> 4) << 3;         // C/D M base: 0 or 8
  const int n0   = wave * 16;                // N tile owned by this wave

  // ---- query tile = last_hidden @ Wq^T + bq  (B[k][n] = Wq[n][k])
  v8f acc = {};
  for (int kk = 0; kk < H_ / 4; ++kk) {
    const int k0 = kk * 4;
    v2f a, b;
    a.x = last_hidden[(m0 + r16) * H_ + k0 + kh];
    a.y = last_hidden[(m0 + r16) * H_ + k0 + kh + 1];
    b.x = Wq[(n0 + r16) * H_ + k0 + kh];
    b.y = Wq[(n0 + r16) * H_ + k0 + kh + 1];
    acc = __builtin_amdgcn_wmma_f32_16x16x4_f32(false, a, false, b,
                                                (short)0, acc, false, false);
  }
  const float bias = bq[n0 + r16];
  for (int r = 0; r < 8; ++r)
    q_lds[mb + r][n0 + r16] = acc[r] + bias;
  __syncthreads();

  // ---- qk tile = query @ Wk  (B[k][n] = Wk[k][n], row-major contiguous)
  v8f acc2 = {};
  for (int kk = 0; kk < H_ / 4; ++kk) {
    const int k0 = kk * 4;
    v2f a, b;
    a.x = q_lds[r16][k0 + kh];
    a.y = q_lds[r16][k0 + kh + 1];
    b.x = Wk[(k0 + kh) * H_ + n0 + r16];
    b.y = Wk[(k0 + kh + 1) * H_ + n0 + r16];
    acc2 = __builtin_amdgcn_wmma_f32_16x16x4_f32(false, a, false, b,
                                                 (short)0, acc2, false, false);
  }
  for (int r = 0; r < 8; ++r)
    qk_out[(size_t)(m0 + mb + r) * H_ + n0 + r16] = acc2[r];

  // ---- qb[b] = query[b,:] . bk   (16 tiny dot products; q_lds is read-only now)
  if (threadIdx.x < 16) {
    float s = 0.f;
    for (int h = 0; h < H_; ++h) s += q_lds[threadIdx.x][h] * bk[h];
    qb_out[m0 + threadIdx.x] = s;
  }
}

// ---------------------------------------------------------------------------
// Kernel 2: fused scores -> masked softmax -> context, one block per batch row.
// The [512,128] f32 tile (256KB) is DMA'd into LDS by the Tensor Data Mover
// with hardware row padding (128 DWORDs data + 4 DWORDs pad -> stride 132),
// so lstm_out is read from HBM exactly once (512MB => ~22us @ 23.3 TB/s).
// ---------------------------------------------------------------------------
__global__ __launch_bounds__(256) void ta_attn_kernel(
    const float* __restrict__ lstm_out,      // [B,L,H]
    const int*   __restrict__ lengths,       // [B]
    const float* __restrict__ qk,            // [B,H]
    const float* __restrict__ qb,            // [B]
    float* __restrict__ out)                 // [B,H]
{
  __shared__ __align__(16) float x[L_][XPAD];    // 512*132*4 = 264 KB
  __shared__ float w[L_];
  __shared__ __align__(16) float qv[H_];
  __shared__ float redm[8];
  __shared__ float reds[8];

  const int tid  = threadIdx.x;
  const int lane = tid & 31;
  const int wid  = tid >> 5;
  const int b    = blockIdx.x;
  const float* __restrict__ src = lstm_out + (size_t)b * L_ * H_;

  if (tid < H_) qv[tid] = qk[(size_t)b * H_ + tid];

  // ---- TDM: one descriptor moves the whole [512,128] tile global->LDS.
  // D# group0: count=1 | lds_addr | global_addr[56:0] | type=2
  // D# group1: data_size=4B, pad_enable, pad_interval=128 DW (code 6),
  //            pad_amount=4 DW (code 3), dim0=128, dim1=512,
  //            tile0=128, tile1=512, dim0_stride=128
  {
    const unsigned           lds_base = (unsigned)(size_t)(void*)&x[0][0];
    const unsigned long long ga       = (unsigned long long)(size_t)src;
    const v4u g0 = { 1u,                                    // count=1 (valid)
                     lds_base,                              // bits 63:32
                     (unsigned)(ga & 0xFFFFFFFFull),        // bits 95:64
                     (unsigned)((ga >> 32) & 0x1FFFFFFull)  // bits 120:96
                       | (2u << 30) };                      // type=2 ("image")
    const v8i g1 = { (int)((2u << 16)        // data_size = 4 bytes
                         | (1u << 20)        // pad_enable
                         | (6u << 22)        // pad_interval: 128 DWORDs
                         | (3u << 25)),      // pad_amount : 4 DWORDs
                     (int)(128u << 16),      // tensor_dim0[15:0] = 128
                     (int)(512u << 16),      // dim0 hi=0 | tensor_dim1[15:0]=512
                     (int)(128u << 16),      // dim1 hi=0 | tile_dim0=128
                     (int)(512u),            // tile_dim1=512 | tile_dim2=0
                     (int)(128u),            // tensor_dim0_stride = 128
                     0, 0 };
    const v4i gz = {};
    if (wid == 0) {                          // one TDM issue per block
#if __clang_major__ >= 23
      const v8i gz8 = {};
      __builtin_amdgcn_tensor_load_to_lds(g0, g1, gz, gz, gz8, 0);
#else
      __builtin_amdgcn_tensor_load_to_lds(g0, g1, gz, gz, 0);
#endif
      __builtin_amdgcn_s_wait_tensorcnt(0);
    }
  }
  __syncthreads();

  // ---- scores: each thread owns l = tid and l = tid+256
  const int   len = lengths[b];
  const float qbv = qb[b];
  float sc[2];
  for (int j = 0; j < 2; ++j) {
    const int l = tid + j * 256;
    const float4* xr = (const float4*)&x[l][0];
    const float4* qr = (const float4*)qv;
    float s = 0.f;
    for (int h4 = 0; h4 < H_ / 4; ++h4) {
      const float4 a = xr[h4], q = qr[h4];
      s += a.x * q.x + a.y * q.y + a.z * q.z + a.w * q.w;
    }
    s = (s + qbv) * 0.08838834764831845f;    // 1/sqrt(128)
    if (l >= len) s = -INFINITY;
    sc[j] = s;
  }

  // ---- softmax: block max
  float m = fmaxf(sc[0], sc[1]);
  for (int off = 16; off > 0; off >>= 1) m = fmaxf(m, __shfl_xor(m, off));
  if (lane == 0) redm[wid] = m;
  __syncthreads();
  m = redm[0];
  for (int i = 1; i < 8; ++i) m = fmaxf(m, redm[i]);

  // exp + block sum (weights left unnormalized; fold 1/sum into the output)
  const float e0 = __expf(sc[0] - m);
  const float e1 = __expf(sc[1] - m);
  w[tid]       = e0;
  w[tid + 256] = e1;
  float s = e0 + e1;
  for (int off = 16; off > 0; off >>= 1) s += __shfl_xor(s, off);
  if (lane == 0) reds[wid] = s;
  __syncthreads();
  s = reds[0];
  for (int i = 1; i < 8; ++i) s += reds[i];
  const float inv = 1.0f / s;

  // ---- context: threads 0-127 own h over l=[0,256), 128-255 over l=[256,512)
  const int h    = tid & (H_ - 1);
  const int half = tid >> 7;
  float acc = 0.f;
  for (int l = half * 256; l < half * 256 + 256; ++l)
    acc = fmaf(w[l], x[l][h], acc);
  __syncthreads();                 // qv reads (scores) are done; safe to reuse
  if (half == 1) qv[h] = acc;
  __syncthreads();
  if (half == 0) out[(size_t)b * H_ + h] = (acc + qv[h]) * inv;
}

// ---------------------------------------------------------------------------
extern "C" void kernel_launch(void* const* d_in, const int* in_sizes, int n_in,
                              void* d_out, int out_size, void* d_ws, size_t ws_size,
                              hipStream_t stream) {
  const float* lstm_out    = (const float*)d_in[0];
  const float* last_hidden = (const float*)d_in[1];
  const int*   lengths     = (const int*)d_in[2];
  const float* Wq          = (const float*)d_in[3];
  const float* bq          = (const float*)d_in[4];
  const float* Wk          = (const float*)d_in[5];
  const float* bk          = (const float*)d_in[6];
  float* out = (float*)d_out;

  // workspace: qk [B*H] f32 (1 MB) + qb [B] f32 (8 KB)
  float* qk_ws = (float*)d_ws;
  float* qb_ws = qk_ws + (size_t)B_ * H_;

  ta_qk_kernel<<<B_ / 16, 256, 0, stream>>>(last_hidden, Wq, bq, Wk, bk,
                                            qk_ws, qb_ws);
  ta_attn_kernel<<<B_, 256, 0, stream>>>(lstm_out, lengths, qk_ws, qb_ws, out);
}